// SAGEConv_18141941859016
// MI455X (gfx1250) — compile-verified
//
#include <hip/hip_runtime.h>

typedef __attribute__((ext_vector_type(2))) float v2f;
typedef __attribute__((ext_vector_type(8))) float v8f;

#define IN_C  256
#define OUT_C 128

// ---------------------------------------------------------------------------
// Phase 1: lin[m,n] = sum_k x[m,k] * W[n,k] + b[n]   via V_WMMA_F32_16X16X4_F32
// One wave (32 lanes) computes a 16x16 tile; 8 waves/block cover OUT_C=128.
// A-frag layout (16x4 f32): lanes 0-15 -> row M=lane, K = k+0,k+1 in v0,v1
//                           lanes 16-31 -> row M=lane-16, K = k+2,k+3
// B-frag layout (4x16 f32): lanes 0-15 -> col N=lane, K = k+0 (v0), k+1 (v1)
//                           lanes 16-31 -> col N=lane-16, K = k+2,k+3
// C/D layout: VGPR v holds M = v + 8*(lane>=16), N = lane&15.
// ---------------------------------------------------------------------------
__global__ void __launch_bounds__(256)
gemm_wmma_f32(const float* __restrict__ x, const float* __restrict__ W,
              const float* __restrict__ bias, float* __restrict__ lin) {
  const int wave = threadIdx.x >> 5;
  const int lane = threadIdx.x & 31;
  const int half = lane >> 4;       // 0 or 1
  const int l16  = lane & 15;
  const int m0   = blockIdx.x * 16; // row tile (N=50000 is a multiple of 16)
  const int n0   = wave * 16;       // col tile (8 waves * 16 = 128 = OUT_C)

  // bias depends only on the output column -> identical in all 8 C VGPRs
  const float bv = bias[n0 + l16];
  v8f c = { bv, bv, bv, bv, bv, bv, bv, bv };

  // per-lane base pointers (already offset by the upper-half K shift)
  const float* __restrict__ arow = x + (size_t)(m0 + l16) * IN_C + 2 * half;
  const float* __restrict__ brow = W + (size_t)(n0 + l16) * IN_C + 2 * half;

#pragma unroll 4
  for (int k = 0; k < IN_C; k += 4) {
    v2f a = *(const v2f*)(arow + k);   // 8B aligned: row base 1KB-aligned + {0,8}
    v2f b = *(const v2f*)(brow + k);
    // 8 args: (neg_a, A, neg_b, B, c_mod, C, reuse_a, reuse_b)
    c = __builtin_amdgcn_wmma_f32_16x16x4_f32(false, a, false, b,
                                              (short)0, c, false, false);
  }

  float* __restrict__ orow = lin + (size_t)m0 * OUT_C + n0 + l16;
#pragma unroll
  for (int v = 0; v < 8; ++v) {
    int m = v + 8 * half;
    orow[(size_t)m * OUT_C] = c[v];
  }
}

// ---------------------------------------------------------------------------
// Phase 0: zero the accumulators (atomics accumulate; must re-init every call)
// ---------------------------------------------------------------------------
__global__ void __launch_bounds__(256)
zero_kernel(float* __restrict__ out, int* __restrict__ counts, int nNodes) {
  int i = blockIdx.x * 256 + threadIdx.x;
  if (i < nNodes * OUT_C) out[i] = 0.0f;
  if (i < nNodes)         counts[i] = 0;
}

// ---------------------------------------------------------------------------
// Phase 2: one wave per edge. Gather lin[src] (L2-resident, float4/lane) and
// scatter-add into out[dst] with hardware f32 atomics; lane 0 bumps the count.
// ---------------------------------------------------------------------------
__global__ void __launch_bounds__(256)
scatter_kernel(const float* __restrict__ lin, const int* __restrict__ srcIdx,
               const int* __restrict__ dstIdx, float* __restrict__ out,
               int* __restrict__ counts, int E) {
  int e = blockIdx.x * 8 + (threadIdx.x >> 5);
  if (e >= E) return;
  int lane = threadIdx.x & 31;
  int s = srcIdx[e];
  int d = dstIdx[e];

  float4 v = *(reinterpret_cast<const float4*>(lin + (size_t)s * OUT_C) + lane);
  float* o = out + (size_t)d * OUT_C + lane * 4;
  unsafeAtomicAdd(o + 0, v.x);
  unsafeAtomicAdd(o + 1, v.y);
  unsafeAtomicAdd(o + 2, v.z);
  unsafeAtomicAdd(o + 3, v.w);
  if (lane == 0) atomicAdd(&counts[d], 1);
}

// ---------------------------------------------------------------------------
// Phase 3: mean = sum / max(count, 1)
// ---------------------------------------------------------------------------
__global__ void __launch_bounds__(256)
normalize_kernel(float* __restrict__ out, const int* __restrict__ counts,
                 int nNodes) {
  int i = blockIdx.x * 256 + threadIdx.x;
  if (i < nNodes * OUT_C) {
    float cnt = (float)counts[i >> 7];   // OUT_C == 128
    out[i] = out[i] / fmaxf(cnt, 1.0f);
  }
}

extern "C" void kernel_launch(void* const* d_in, const int* in_sizes, int n_in,
                              void* d_out, int out_size, void* d_ws, size_t ws_size,
                              hipStream_t stream) {
  const float* x    = (const float*)d_in[0];
  const int*   eidx = (const int*)d_in[1];   // [2, E] int32
  const float* W    = (const float*)d_in[2]; // [OUT_C, IN_C] row-major
  const float* bias = (const float*)d_in[3]; // [OUT_C]
  float*       out  = (float*)d_out;

  const int nNodes = in_sizes[0] / IN_C;  // 50000
  const int E      = in_sizes[1] / 2;     // 800000
  const int* srcIdx = eidx;
  const int* dstIdx = eidx + E;

  // workspace: [counts: nNodes ints][lin: nNodes*OUT_C floats], 16B-aligned
  int*   counts = (int*)d_ws;
  size_t cntPad = (size_t)((nNodes + 3) & ~3);
  float* lin    = (float*)d_ws + cntPad;

  const int total = nNodes * OUT_C;
  zero_kernel     <<<(total + 255) / 256, 256, 0, stream>>>(out, counts, nNodes);
  gemm_wmma_f32   <<<nNodes / 16,         256, 0, stream>>>(x, W, bias, lin);
  scatter_kernel  <<<(E + 7) / 8,         256, 0, stream>>>(lin, srcIdx, dstIdx,
                                                            out, counts, E);
  normalize_kernel<<<(total + 255) / 256, 256, 0, stream>>>(out, counts, nNodes);
}